// LightGCN_52286931862209
// MI455X (gfx1250) — compile-verified
//
#include <hip/hip_runtime.h>
#include <math.h>

#define NUM_USERS 100000
#define NUM_ITEMS 50000
#define NNODES    150000
#define DIM       64
#define NUM_EDGES 1000000
#define BATCH     16384
#define NELEM     (NNODES * DIM)   /* 9,600,000 */

typedef float v2f __attribute__((ext_vector_type(2)));
typedef float v8f __attribute__((ext_vector_type(8)));

// ---------------------------------------------------------------------------
// init: cur = concat(user_emb, item_emb); acc (= out+1 region) = 0.25*cur;
//       out[0] (loss) = 0
// ---------------------------------------------------------------------------
__global__ void lgcn_init(const float* __restrict__ ue, const float* __restrict__ ie,
                          float* __restrict__ cur, float* __restrict__ out) {
  long i = (long)blockIdx.x * blockDim.x + threadIdx.x;
  if (i == 0) out[0] = 0.0f;
  if (i >= NELEM) return;
  float e = (i < (long)NUM_USERS * DIM) ? ue[i] : ie[i - (long)NUM_USERS * DIM];
  cur[i] = e;
  out[1 + i] = 0.25f * e;
}

__global__ void lgcn_zero(float* __restrict__ p) {
  long i = (long)blockIdx.x * blockDim.x + threadIdx.x;
  if (i < NELEM) p[i] = 0.0f;
}

// ---------------------------------------------------------------------------
// SpMM scatter: one wave32 per edge; lane l handles dims 2l, 2l+1.
// Gather cur[col] (L2-resident), atomic-scatter val*cur[col] into nxt[row].
// ---------------------------------------------------------------------------
__global__ void lgcn_spmm(const int* __restrict__ erow, const int* __restrict__ ecol,
                          const float* __restrict__ eval, const float* __restrict__ cur,
                          float* __restrict__ nxt) {
  int wid  = (blockIdx.x * blockDim.x + threadIdx.x) >> 5;
  int lane = threadIdx.x & 31;
  if (wid >= NUM_EDGES) return;
  int r = erow[wid];
  int c = ecol[wid];
  float v = eval[wid];
  const float2* src = (const float2*)(cur + (long)c * DIM);
  float2 x = src[lane];
  float* dst = nxt + (long)r * DIM + lane * 2;
  atomicAdd(dst,     v * x.x);
  atomicAdd(dst + 1, v * x.y);
}

// ---------------------------------------------------------------------------
// Row L2-normalize in place (wave32 per row, shfl_xor reduction) and
// accumulate 0.25 * normalized row into the output embedding region.
// ---------------------------------------------------------------------------
__global__ void lgcn_norm_acc(float* __restrict__ buf, float* __restrict__ out) {
  int row  = (blockIdx.x * blockDim.x + threadIdx.x) >> 5;
  int lane = threadIdx.x & 31;
  if (row >= NNODES) return;
  float2* p = (float2*)(buf + (long)row * DIM);
  float2 x = p[lane];
  float ss = x.x * x.x + x.y * x.y;
#pragma unroll
  for (int o = 16; o > 0; o >>= 1) ss += __shfl_xor(ss, o, 32);
  float scale = 1.0f / fmaxf(sqrtf(ss), 1e-12f);
  x.x *= scale;
  x.y *= scale;
  p[lane] = x;
  float* a = out + 1 + (long)row * DIM + lane * 2;   // 4B-aligned scalar access
  a[0] += 0.25f * x.x;
  a[1] += 0.25f * x.y;
}

// ---------------------------------------------------------------------------
// BPR scoring via V_WMMA_F32_16X16X4_F32: one wave per 16 batch rows.
// S_pos = U(16x64) * P^T, S_neg = U * N^T, accumulated as 16 chained
// 16x16x4 f32 WMMAs (exact f32 math). Diagonal -> stable softplus ->
// wave reduction -> atomicAdd(loss, sum/BATCH).
//
// A-fragment layout (ISA 16x4 f32): lanes 0-15 hold {K+0,K+1} of row M=lane,
// lanes 16-31 hold {K+2,K+3} of row M=lane-16. The B-fragment of a
// row-major matrix's transpose uses the same per-lane addressing.
// ---------------------------------------------------------------------------
__global__ void lgcn_score(const float* __restrict__ emb /* = out+1 */,
                           const int* __restrict__ uid, const int* __restrict__ pid,
                           const int* __restrict__ nid, float* __restrict__ loss) {
  int wid  = (blockIdx.x * blockDim.x + threadIdx.x) >> 5;
  int lane = threadIdx.x & 31;
  int base = wid * 16;                 // grid sized exactly: BATCH/16 waves
  int m    = lane & 15;
  int koff = (lane < 16) ? 0 : 2;

  const float* urow = emb + (long)uid[base + m] * DIM;
  const float* prow = emb + ((long)pid[base + m] + NUM_USERS) * DIM;
  const float* nrow = emb + ((long)nid[base + m] + NUM_USERS) * DIM;

  v8f cp = {0.f, 0.f, 0.f, 0.f, 0.f, 0.f, 0.f, 0.f};
  v8f cn = {0.f, 0.f, 0.f, 0.f, 0.f, 0.f, 0.f, 0.f};

#pragma unroll
  for (int k = 0; k < DIM; k += 4) {
    v2f a, bp, bn;
    a.x  = urow[k + koff];  a.y  = urow[k + koff + 1];
    bp.x = prow[k + koff];  bp.y = prow[k + koff + 1];
    bn.x = nrow[k + koff];  bn.y = nrow[k + koff + 1];
    cp = __builtin_amdgcn_wmma_f32_16x16x4_f32(false, a, false, bp, (short)0, cp,
                                               false, false);
    cn = __builtin_amdgcn_wmma_f32_16x16x4_f32(false, a, false, bn, (short)0, cn,
                                               false, false);
  }

  // Diagonal: VGPR r holds M=r (lanes 0-15, N=lane) and M=r+8 (lanes 16-31,
  // N=lane-16).  M==N  =>  lane==r  or  lane==r+24.
  float psum = 0.0f;
#pragma unroll
  for (int r = 0; r < 8; ++r) {
    if (lane == r || lane == r + 24) {
      float z = -(cp[r] - cn[r]);                       // softplus(-(pos-neg))
      psum += fmaxf(z, 0.0f) + log1pf(expf(-fabsf(z))); // numerically stable
    }
  }
#pragma unroll
  for (int o = 16; o > 0; o >>= 1) psum += __shfl_xor(psum, o, 32);
  if (lane == 0) atomicAdd(loss, psum * (1.0f / (float)BATCH));
}

// ---------------------------------------------------------------------------
extern "C" void kernel_launch(void* const* d_in, const int* in_sizes, int n_in,
                              void* d_out, int out_size, void* d_ws, size_t ws_size,
                              hipStream_t stream) {
  (void)in_sizes; (void)n_in; (void)out_size; (void)ws_size;
  const float* ue   = (const float*)d_in[0];
  const float* ie   = (const float*)d_in[1];
  const int*   erow = (const int*)d_in[2];
  const int*   ecol = (const int*)d_in[3];
  const float* eval = (const float*)d_in[4];
  const int*   uid  = (const int*)d_in[5];
  const int*   pid  = (const int*)d_in[6];
  const int*   nid  = (const int*)d_in[7];
  float* out  = (float*)d_out;
  float* buf0 = (float*)d_ws;             // ping
  float* buf1 = buf0 + NELEM;             // pong (ws >= 76.8 MB)

  dim3 blk(256);
  const int gElem  = (NELEM + 255) / 256;          // 37500 blocks
  const int gSpmm  = (NUM_EDGES * 32 + 255) / 256; // 125000 blocks (wave/edge)
  const int gNorm  = (NNODES * 32 + 255) / 256;    // 18750 blocks (wave/row)
  const int gScore = (BATCH / 16) * 32 / 256;      // 128 blocks, all waves full

  lgcn_init<<<gElem, blk, 0, stream>>>(ue, ie, buf0, out);

  float* cur = buf0;
  float* nxt = buf1;
  for (int layer = 0; layer < 3; ++layer) {
    lgcn_zero<<<gElem, blk, 0, stream>>>(nxt);
    lgcn_spmm<<<gSpmm, blk, 0, stream>>>(erow, ecol, eval, cur, nxt);
    lgcn_norm_acc<<<gNorm, blk, 0, stream>>>(nxt, out);
    float* t = cur; cur = nxt; nxt = t;
  }

  lgcn_score<<<gScore, blk, 0, stream>>>(out + 1, uid, pid, nid, out);
}